// NetVLAD_7395933684184
// MI455X (gfx1250) — compile-verified
//
#include <hip/hip_runtime.h>
#include <hip/hip_bf16.h>

typedef float v2f __attribute__((ext_vector_type(2)));
typedef float v8f __attribute__((ext_vector_type(8)));
typedef unsigned int u32x4 __attribute__((ext_vector_type(4)));
typedef int i32x4 __attribute__((ext_vector_type(4)));
typedef int i32x8 __attribute__((ext_vector_type(8)));

#define N_IMG   64
#define C_DIM   128
#define K_CL    64
#define P_PIX   4096
#define SLICES  8
#define P_SLICE 512
#define TP      64          // pixels per inner tile
#define NTILES  (P_SLICE / TP)

// LDS layout (float offsets); stride 68 floats = TDM pad (64 DW + 4 DW pad)
#define W_STR   132
#define X_STR   68
#define A_STR   68
#define W_OFF   0
#define B_OFF   (W_OFF + K_CL * W_STR)        // 8448
#define X_OFF   (B_OFF + 64)                  // 8512
#define XBUF_F  (C_DIM * X_STR)               // 8704 floats per x buffer
#define A_OFF   (X_OFF + 2 * XBUF_F)          // 25920
#define SMEM_FLOATS (A_OFF + K_CL * A_STR)    // 30272 -> 121088 bytes

// ---------------------------------------------------------------------------
// TDM: DMA one 2-D x-tile (tile_dim0=64 pixels contiguous, tile_dim1=128
// channels, tensor_dim0_stride=4096) from global into LDS with pad to
// stride-68 rows. Descriptor per CDNA5 ISA.
// ---------------------------------------------------------------------------
__device__ __forceinline__ void tdm_load_x_tile(const float* gsrc, unsigned lds_byte)
{
    unsigned long long ga = (unsigned long long)gsrc;

    u32x4 g0;
    g0[0] = 1u;                                          // count=1, user D#
    g0[1] = lds_byte;                                    // lds_addr [63:32]
    g0[2] = (unsigned)(ga & 0xFFFFFFFFull);              // global_addr[31:0]
    g0[3] = (unsigned)((ga >> 32) & 0x01FFFFFFull)       // global_addr[56:32]
          | 0x80000000u;                                 // type=2 ("image")

    i32x8 g1;
    // data_size=2 (4B) | pad_enable | pad_interval=5 (64 DW) | pad_amount=3 (4 DW)
    g1[0] = (int)((2u << 16) | (1u << 20) | (5u << 22) | (3u << 25));
    g1[1] = (int)(P_PIX << 16);     // tensor_dim0 = 4096 (bits [79:48], low 16)
    g1[2] = (int)(C_DIM << 16);     // tensor_dim0 hi = 0; tensor_dim1 = 128
    g1[3] = (int)(TP << 16);        // tensor_dim1 hi = 0; tile_dim0 = 64
    g1[4] = C_DIM;                  // tile_dim1 = 128; tile_dim2 = 0
    g1[5] = P_PIX;                  // tensor_dim0_stride = 4096 (low 32)
    g1[6] = 0;
    g1[7] = 0;

    i32x4 gz;
    gz[0] = 0; gz[1] = 0; gz[2] = 0; gz[3] = 0;

#if defined(__clang_major__) && (__clang_major__ >= 23)
    i32x8 gz8;
    gz8[0] = 0; gz8[1] = 0; gz8[2] = 0; gz8[3] = 0;
    gz8[4] = 0; gz8[5] = 0; gz8[6] = 0; gz8[7] = 0;
    __builtin_amdgcn_tensor_load_to_lds(g0, g1, gz, gz, gz8, 0);
#else
    __builtin_amdgcn_tensor_load_to_lds(g0, g1, gz, gz, 0);
#endif
}

__global__ void vlad_accum(const float* __restrict__ x,
                           const float* __restrict__ cw,
                           const float* __restrict__ cb,
                           float* __restrict__ ws_vlad,
                           float* __restrict__ ws_asum)
{
    extern __shared__ float sm[];
    float* w_lds = sm + W_OFF;
    float* b_lds = sm + B_OFF;
    float* x_lds = sm + X_OFF;   // two buffers of XBUF_F floats
    float* a_lds = sm + A_OFF;

    const int tid  = threadIdx.x;
    const int lane = tid & 31;
    const int wave = tid >> 5;
    const int half = lane >> 4;   // 0: lanes 0-15, 1: lanes 16-31
    const int l    = lane & 15;

    const int n  = blockIdx.x >> 3;      // image
    const int s  = blockIdx.x & 7;       // P-slice
    const int p0 = s * P_SLICE;

    const float* xbase = x + (size_t)n * C_DIM * P_PIX;
    const unsigned xlds_byte0 = (unsigned)(unsigned long long)(x_lds);
    const unsigned xlds_byte1 = (unsigned)(unsigned long long)(x_lds + XBUF_F);

    // stage conv_w (64x128) into LDS, padded stride (done once, cached all slices)
    for (int i = tid; i < (K_CL * C_DIM) / 4; i += 256) {
        int flat = i * 4;
        int row  = flat >> 7;
        int col  = flat & 127;
        float4 v = *(const float4*)(cw + flat);
        *(float4*)(w_lds + row * W_STR + col) = v;
    }
    if (tid < K_CL) b_lds[tid] = cb[tid];

    // persistent GEMM2 accumulators: wave owns k rows k0..k0+15, 4 col-tiles
    const int k0     = 16 * (wave & 3);
    const int c0base = 16 * (wave >> 2);
    v8f acc2[4];
    for (int j = 0; j < 4; ++j)
        for (int r = 0; r < 8; ++r) acc2[j][r] = 0.f;
    float asum_reg = 0.f;

    // prologue: kick off DMA of tile 0 into buffer 0 (wave 0 owns the TDM)
    if (wave == 0) {
        tdm_load_x_tile(xbase + p0, xlds_byte0);
    }

    for (int t = 0; t < NTILES; ++t) {
        const int cur = t & 1;
        __syncthreads();   // all waves done reading buf[1-cur] / a_lds from t-1

        if (wave == 0) {
            if (t + 1 < NTILES) {
                // prefetch next tile into the other buffer while we compute
                tdm_load_x_tile(xbase + p0 + (t + 1) * TP,
                                cur ? xlds_byte0 : xlds_byte1);
                __builtin_amdgcn_s_wait_tensorcnt(1);  // tile t (older) done
            } else {
                __builtin_amdgcn_s_wait_tensorcnt(0);  // last tile done
            }
        }
        __syncthreads();   // publish buf[cur] to all waves
        const float* xc = x_lds + cur * XBUF_F;

        // GEMM1: logits(64x64) = conv_w(64x128) * xtile(128x64); 2 tiles/wave
        for (int u = 0; u < 2; ++u) {
            int tt  = 2 * wave + u;
            int m0  = 16 * (tt & 3);
            int pp0 = 16 * (tt >> 2);
            v8f acc;
            for (int r = 0; r < 8; ++r) acc[r] = 0.f;
            #pragma unroll 8
            for (int cs = 0; cs < 32; ++cs) {
                int c4 = 4 * cs;
                v2f af = *(const v2f*)(w_lds + (m0 + l) * W_STR + c4 + 2 * half);
                int xrow = c4 + 2 * half;
                v2f bf;
                bf.x = xc[(xrow + 0) * X_STR + pp0 + l];   // fuses to ds_load_2addr
                bf.y = xc[(xrow + 1) * X_STR + pp0 + l];
                acc = __builtin_amdgcn_wmma_f32_16x16x4_f32(
                          false, af, false, bf, (short)0, acc, false, false);
            }
            #pragma unroll
            for (int r = 0; r < 8; ++r) {
                int m = m0 + r + 8 * half;
                a_lds[m * A_STR + pp0 + l] = acc[r] + b_lds[m];
            }
        }
        __syncthreads();

        // softmax over K per pixel: 64 pixels -> waves 0,1 (no intra-wave div.)
        if (tid < TP) {
            int p = tid;
            float mx = -3.0e38f;
            for (int k = 0; k < K_CL; ++k) mx = fmaxf(mx, a_lds[k * A_STR + p]);
            float ssum = 0.f;
            for (int k = 0; k < K_CL; ++k) {
                float e = __expf(a_lds[k * A_STR + p] - mx);
                a_lds[k * A_STR + p] = e;
                ssum += e;
            }
            float inv = 1.f / ssum;
            for (int k = 0; k < K_CL; ++k) a_lds[k * A_STR + p] *= inv;
        }
        __syncthreads();

        // asum[k] partials (read-only on a_lds; runs alongside GEMM2)
        if (tid < K_CL) {
            float s2 = 0.f;
            for (int p = 0; p < TP; ++p) s2 += a_lds[tid * A_STR + p];
            asum_reg += s2;
        }

        // GEMM2: vlad(64x128) += a(64x64) * x^T(64x128)
        for (int j = 0; j < 4; ++j) {
            int c0 = c0base + 32 * j;
            #pragma unroll 8
            for (int ks = 0; ks < 16; ++ks) {
                int pk = 4 * ks;
                v2f af = *(const v2f*)(a_lds + (k0 + l) * A_STR + pk + 2 * half);
                v2f bf = *(const v2f*)(xc    + (c0 + l) * X_STR + pk + 2 * half);
                acc2[j] = __builtin_amdgcn_wmma_f32_16x16x4_f32(
                              false, af, false, bf, (short)0, acc2[j], false, false);
            }
        }
    }

    // flush per-slice accumulators into workspace
    float* vbase = ws_vlad + (size_t)n * K_CL * C_DIM;
    for (int j = 0; j < 4; ++j) {
        int c0 = c0base + 32 * j;
        #pragma unroll
        for (int r = 0; r < 8; ++r) {
            int k = k0 + r + 8 * half;
            unsafeAtomicAdd(vbase + k * C_DIM + c0 + l, acc2[j][r]);
        }
    }
    if (tid < K_CL) unsafeAtomicAdd(ws_asum + n * K_CL + tid, asum_reg);
}

__global__ void vlad_finalize(const float* __restrict__ ws_vlad,
                              const float* __restrict__ ws_asum,
                              const float* __restrict__ cent,
                              float* __restrict__ out)
{
    __shared__ float v_lds[K_CL * 129];
    __shared__ float a_s[K_CL];
    __shared__ float rsc[K_CL];
    __shared__ float pr[256];
    __shared__ float ginv;

    const int n   = blockIdx.x;
    const int tid = threadIdx.x;

    if (tid < K_CL) a_s[tid] = ws_asum[n * K_CL + tid];
    __syncthreads();

    // v = vlad - asum[k] * centroids[k][c]
    for (int i = tid; i < K_CL * C_DIM; i += 256) {
        int k = i >> 7, c = i & 127;
        float v = ws_vlad[(size_t)n * K_CL * C_DIM + i] - a_s[k] * cent[k * C_DIM + c];
        v_lds[k * 129 + c] = v;
    }
    __syncthreads();

    // per-cluster L2 norm
    if (tid < K_CL) {
        float sq = 0.f;
        for (int c = 0; c < C_DIM; ++c) { float tv = v_lds[tid * 129 + c]; sq += tv * tv; }
        rsc[tid] = 1.f / fmaxf(sqrtf(sq), 1e-12f);
    }
    __syncthreads();

    // global L2 norm over all 8192 normalized values
    float part = 0.f;
    for (int i = tid; i < K_CL * C_DIM; i += 256) {
        int k = i >> 7, c = i & 127;
        float tv = v_lds[k * 129 + c] * rsc[k];
        part += tv * tv;
    }
    pr[tid] = part;
    __syncthreads();
    if (tid == 0) {
        float S = 0.f;
        for (int i = 0; i < 256; ++i) S += pr[i];
        ginv = 1.f / fmaxf(sqrtf(S), 1e-12f);
    }
    __syncthreads();

    for (int i = tid; i < K_CL * C_DIM; i += 256) {
        int k = i >> 7, c = i & 127;
        out[(size_t)n * K_CL * C_DIM + i] = v_lds[k * 129 + c] * rsc[k] * ginv;
    }
}

extern "C" void kernel_launch(void* const* d_in, const int* in_sizes, int n_in,
                              void* d_out, int out_size, void* d_ws, size_t ws_size,
                              hipStream_t stream) {
    const float* x    = (const float*)d_in[0];   // (64,128,64,64)
    const float* cw   = (const float*)d_in[1];   // (64,128)
    const float* cb   = (const float*)d_in[2];   // (64,)
    const float* cent = (const float*)d_in[3];   // (64,128)
    float* out = (float*)d_out;                  // (64, 8192)

    float* ws_vlad = (float*)d_ws;                                   // N*K*C f32
    float* ws_asum = ws_vlad + (size_t)N_IMG * K_CL * C_DIM;         // N*K f32
    size_t zero_bytes = ((size_t)N_IMG * K_CL * C_DIM + (size_t)N_IMG * K_CL) * sizeof(float);
    hipMemsetAsync(d_ws, 0, zero_bytes, stream);

    hipLaunchKernelGGL(vlad_accum, dim3(N_IMG * SLICES), dim3(256),
                       SMEM_FLOATS * sizeof(float), stream,
                       x, cw, cb, ws_vlad, ws_asum);
    hipLaunchKernelGGL(vlad_finalize, dim3(N_IMG), dim3(256), 0, stream,
                       ws_vlad, ws_asum, cent, out);
}